// MetaAdapter_19224273617418
// MI455X (gfx1250) — compile-verified
//
#include <hip/hip_runtime.h>
#include <math.h>

// MetaAdapter on MI455X (gfx1250, wave32).
// Restructured: Wc = Wk^T Wq (wmma f32), T = Q Wc^T (wmma f32),
// then a streaming softmax/gate/normalize kernel bound by key+value HBM traffic.

typedef float v2f __attribute__((ext_vector_type(2)));
typedef float v8f __attribute__((ext_vector_type(8)));

#define DIM 1024
#define NB  8192
#define NK  16

// ---------------------------------------------------------------------------
// Kernel 1: Wc[i][j] = sum_c Wk[c][i] * Wq[c][j]   (Wc = Wk^T @ Wq)
// One wave per 16x16 tile; 64x64 tiles; 8 waves/block -> 512 blocks.
// ---------------------------------------------------------------------------
__global__ void __launch_bounds__(256) wc_gemm_kernel(const float* __restrict__ Wq,
                                                      const float* __restrict__ Wk,
                                                      float* __restrict__ wc) {
    const int lane = threadIdx.x & 31;
    const int wave = threadIdx.x >> 5;
    const int tile = blockIdx.x * 8 + wave;      // 4096 tiles
    const int m0 = (tile >> 6) * 16;             // i tile
    const int n0 = (tile & 63) * 16;             // j tile
    const int half = lane >> 4;                  // 0 or 1
    const int lm   = lane & 15;

    v8f acc = {};
    for (int k0 = 0; k0 < DIM; k0 += 4) {
        const int kb = k0 + 2 * half;
        v2f a, b;
        // A[m][c] = Wk[c][m] (c-strided, lane-contiguous in m)
        a.x = Wk[kb * DIM + m0 + lm];
        a.y = Wk[(kb + 1) * DIM + m0 + lm];
        // B[c][n] = Wq[c][n]
        b.x = Wq[kb * DIM + n0 + lm];
        b.y = Wq[(kb + 1) * DIM + n0 + lm];
        acc = __builtin_amdgcn_wmma_f32_16x16x4_f32(false, a, false, b,
                                                    (short)0, acc, false, false);
    }
#pragma unroll
    for (int v = 0; v < 8; ++v)
        wc[(m0 + v + 8 * half) * DIM + (n0 + lm)] = acc[v];
}

// ---------------------------------------------------------------------------
// Kernel 2: T[b][i] = sum_j Q[b][j] * Wc[i][j]    (T = Q @ Wc^T)
// One wave per 16(M) x 64(N) strip: A fragment reused across 4 WMMAs.
// 512 M-tiles * 16 N-groups = 8192 waves -> 1024 blocks.
// ---------------------------------------------------------------------------
__global__ void __launch_bounds__(256) t_gemm_kernel(const float* __restrict__ query,
                                                     const float* __restrict__ wc,
                                                     float* __restrict__ T) {
    const int lane  = threadIdx.x & 31;
    const int wave  = threadIdx.x >> 5;
    const int wt    = blockIdx.x * 8 + wave;     // 8192 wave strips
    const int m0    = (wt >> 4) * 16;            // batch-row tile
    const int n0    = (wt & 15) * 64;            // output-col group
    const int half  = lane >> 4;
    const int lm    = lane & 15;

    v8f a0 = {}, a1 = {}, a2 = {}, a3 = {};
    const float* arow = query + (size_t)(m0 + lm) * DIM;
    const float* b0r  = wc + (size_t)(n0 +  0 + lm) * DIM;
    const float* b1r  = wc + (size_t)(n0 + 16 + lm) * DIM;
    const float* b2r  = wc + (size_t)(n0 + 32 + lm) * DIM;
    const float* b3r  = wc + (size_t)(n0 + 48 + lm) * DIM;

    for (int k0 = 0; k0 < DIM; k0 += 4) {
        const int kb = k0 + 2 * half;
        v2f a  = *(const v2f*)(arow + kb);       // A[m][k] = Q[b][j], contiguous in k
        v2f b0 = *(const v2f*)(b0r + kb);        // B[k][n] = Wc[n][k], contiguous in k
        v2f b1 = *(const v2f*)(b1r + kb);
        v2f b2 = *(const v2f*)(b2r + kb);
        v2f b3 = *(const v2f*)(b3r + kb);
        a0 = __builtin_amdgcn_wmma_f32_16x16x4_f32(false, a, false, b0, (short)0, a0, false, false);
        a1 = __builtin_amdgcn_wmma_f32_16x16x4_f32(false, a, false, b1, (short)0, a1, false, false);
        a2 = __builtin_amdgcn_wmma_f32_16x16x4_f32(false, a, false, b2, (short)0, a2, false, false);
        a3 = __builtin_amdgcn_wmma_f32_16x16x4_f32(false, a, false, b3, (short)0, a3, false, false);
    }
#pragma unroll
    for (int v = 0; v < 8; ++v) {
        const size_t row = (size_t)(m0 + v + 8 * half) * DIM;
        T[row + n0 +  0 + lm] = a0[v];
        T[row + n0 + 16 + lm] = a1[v];
        T[row + n0 + 32 + lm] = a2[v];
        T[row + n0 + 48 + lm] = a3[v];
    }
}

// ---------------------------------------------------------------------------
// Kernel 3: streaming attention epilogue. One 256-thread block per batch row.
// scores[0]=t.q, scores[1+j]=t.key_j ; softmax(scale=1/32) ; gate ; normalize.
// ---------------------------------------------------------------------------
__global__ void __launch_bounds__(256) attn_kernel(const float* __restrict__ query,
                                                   const float* __restrict__ key,
                                                   const float* __restrict__ value,
                                                   const float* __restrict__ Wa,
                                                   const float* __restrict__ ba,
                                                   const float* __restrict__ T,
                                                   float* __restrict__ out) {
    const int b    = blockIdx.x;
    const int tid  = threadIdx.x;
    const int d0   = tid * 4;
    const int wave = tid >> 5;
    const int lane = tid & 31;

    const float4 q4  = *(const float4*)(query + (size_t)b * DIM + d0);
    const float4 t4  = *(const float4*)(T     + (size_t)b * DIM + d0);
    const float4 wa4 = *(const float4*)(Wa + d0);

    float p[18];
    p[0]  = t4.x * q4.x + t4.y * q4.y + t4.z * q4.z + t4.w * q4.w;
    p[17] = q4.x * wa4.x + q4.y * wa4.y + q4.z * wa4.z + q4.w * wa4.w;
    const float* krow = key + (size_t)b * NK * DIM + d0;
#pragma unroll
    for (int j = 0; j < NK; ++j) {
        float4 k4 = *(const float4*)(krow + j * DIM);
        p[1 + j] = t4.x * k4.x + t4.y * k4.y + t4.z * k4.z + t4.w * k4.w;
    }

    // wave32 tree reduction of all 18 partials
#pragma unroll
    for (int i = 0; i < 18; ++i)
#pragma unroll
        for (int off = 16; off > 0; off >>= 1)
            p[i] += __shfl_xor(p[i], off, 32);

    __shared__ float red[8][18];
    __shared__ float fin[18];
    __shared__ float nred[8];
    if (lane == 0) {
#pragma unroll
        for (int i = 0; i < 18; ++i) red[wave][i] = p[i];
    }
    __syncthreads();
    if (tid < 18) {
        float s = 0.f;
#pragma unroll
        for (int w = 0; w < 8; ++w) s += red[w][tid];
        fin[tid] = s;
    }
    __syncthreads();

    // softmax over 17 scores (scale = 1/sqrt(1024) = 1/32), redundantly per thread
    const float scale = 0.03125f;
    float s[17], mx = -1e30f;
#pragma unroll
    for (int i = 0; i < 17; ++i) { s[i] = fin[i] * scale; mx = fmaxf(mx, s[i]); }
    float sum = 0.f;
#pragma unroll
    for (int i = 0; i < 17; ++i) { s[i] = __expf(s[i] - mx); sum += s[i]; }
    const float isum  = 1.0f / sum;
    const float alpha = 1.0f / (1.0f + __expf(-(fin[17] + ba[0])));

    // attn = w0*q + sum_j w_{1+j} * value_j
    float w0 = s[0] * isum;
    float ax = w0 * q4.x, ay = w0 * q4.y, az = w0 * q4.z, aw = w0 * q4.w;
    const float* vrow = value + (size_t)b * NK * DIM + d0;
#pragma unroll
    for (int j = 0; j < NK; ++j) {
        float4 v4 = *(const float4*)(vrow + j * DIM);
        float wj = s[1 + j] * isum;
        ax += wj * v4.x; ay += wj * v4.y; az += wj * v4.z; aw += wj * v4.w;
    }
    float ox = q4.x + alpha * ax;
    float oy = q4.y + alpha * ay;
    float oz = q4.z + alpha * az;
    float ow = q4.w + alpha * aw;

    // L2 norm
    float np = ox * ox + oy * oy + oz * oz + ow * ow;
#pragma unroll
    for (int off = 16; off > 0; off >>= 1) np += __shfl_xor(np, off, 32);
    if (lane == 0) nred[wave] = np;
    __syncthreads();
    float tot = 0.f;
#pragma unroll
    for (int w = 0; w < 8; ++w) tot += nred[w];
    const float invn = 1.0f / fmaxf(sqrtf(tot), 1e-12f);

    float4 res = make_float4(ox * invn, oy * invn, oz * invn, ow * invn);
    *(float4*)(out + (size_t)b * DIM + d0) = res;
}

// ---------------------------------------------------------------------------
extern "C" void kernel_launch(void* const* d_in, const int* in_sizes, int n_in,
                              void* d_out, int out_size, void* d_ws, size_t ws_size,
                              hipStream_t stream) {
    const float* query = (const float*)d_in[0];
    const float* key   = (const float*)d_in[1];
    const float* value = (const float*)d_in[2];
    const float* Wq    = (const float*)d_in[3];
    const float* Wk    = (const float*)d_in[4];
    const float* Wa    = (const float*)d_in[5];
    const float* ba    = (const float*)d_in[6];
    float* out = (float*)d_out;

    float* wc = (float*)d_ws;                 // 1024*1024 floats = 4 MB
    float* T  = wc + (size_t)DIM * DIM;       // 8192*1024 floats = 33.5 MB

    wc_gemm_kernel<<<512, 256, 0, stream>>>(Wq, Wk, wc);
    t_gemm_kernel<<<1024, 256, 0, stream>>>(query, wc, T);
    attn_kernel<<<NB, 256, 0, stream>>>(query, key, value, Wa, ba, T, out);
}